// LearnedWLSSetTransformer_2671469658618
// MI455X (gfx1250) — compile-verified
//
#include <hip/hip_runtime.h>

typedef float v2f __attribute__((ext_vector_type(2)));
typedef float v8f __attribute__((ext_vector_type(8)));

#define BDIM   2048
#define MTOK   1024
#define DDIM   64
#define TB     16        // batches per workgroup
#define NEGINF (-1e30f)

__global__ __launch_bounds__(512) void wls_settf_fused(
    const float* __restrict__ x,          // (B, M, 4)
    const unsigned char* __restrict__ pad,// (B, M) bool
    const float* __restrict__ W_in,       // (4, 64)
    const float* __restrict__ b_in,       // (64)
    const float* __restrict__ seed,       // (64)
    const float* __restrict__ Wq,         // (64,64)
    const float* __restrict__ Wk,         // (64,64)
    const float* __restrict__ Wv,         // (64,64)
    const float* __restrict__ Wo,         // (64,64)
    const float* __restrict__ b_o,        // (64)
    const float* __restrict__ w_loc,      // (68)
    const float* __restrict__ b_loc,      // (1)
    const float* __restrict__ prelu_a,    // (1)
    float* __restrict__ out)              // (B, 4)
{
    __shared__ float s_q[DDIM];
    __shared__ float s_qk[DDIM];
    __shared__ float s_u[4];
    __shared__ float s_sc;
    __shared__ float s_xbar[TB][4];
    __shared__ float s_hbar[TB][DDIM];
    __shared__ float s_pool[TB][DDIM];
    __shared__ float s_feat[TB][DDIM];
    __shared__ float s_g[TB];

    const int tid  = threadIdx.x;
    const int wave = tid >> 5;
    const int lane = tid & 31;
    const int b    = blockIdx.x * TB + wave;   // batch handled by this wave

    // ---------------- Phase 0: q = seed@Wq, qk = Wk@q, u = W_in@qk ------------
    if (tid < DDIM) {
        float acc = 0.f;
        for (int e = 0; e < DDIM; ++e) acc += seed[e] * Wq[e * DDIM + tid];
        s_q[tid] = acc;
    }
    __syncthreads();
    if (tid < DDIM) {
        float acc = 0.f;
        for (int d = 0; d < DDIM; ++d) acc += Wk[tid * DDIM + d] * s_q[d];
        s_qk[tid] = acc;
    }
    __syncthreads();
    if (tid < 4) {
        float acc = 0.f;
        for (int e = 0; e < DDIM; ++e) acc += W_in[tid * DDIM + e] * s_qk[e];
        s_u[tid] = acc * 0.125f;               // fold 1/sqrt(64)
    }
    if (tid == 4) {
        float acc = 0.f;
        for (int e = 0; e < DDIM; ++e) acc += b_in[e] * s_qk[e];
        s_sc = acc * 0.125f;
    }
    __syncthreads();

    // ---------------- Phase 1: online masked softmax -> xbar ------------------
    {
        const float u0 = s_u[0], u1 = s_u[1], u2 = s_u[2], u3 = s_u[3], sc = s_sc;
        const float4* xb = reinterpret_cast<const float4*>(x + (size_t)b * MTOK * 4);
        const unsigned char* pm = pad + (size_t)b * MTOK;

        float mx = -INFINITY, S = 0.f, P0 = 0.f, P1 = 0.f, P2 = 0.f, P3 = 0.f;
        for (int m = lane; m < MTOK; m += 32) {
            float4 xv = xb[m];
            float s = xv.x * u0 + xv.y * u1 + xv.z * u2 + xv.w * u3 + sc;
            if (pm[m]) s = NEGINF;
            if (s > mx) {
                float r = expf(mx - s);
                S  = S  * r + 1.f;
                P0 = P0 * r + xv.x;  P1 = P1 * r + xv.y;
                P2 = P2 * r + xv.z;  P3 = P3 * r + xv.w;
                mx = s;
            } else {
                float e = expf(s - mx);
                S += e;
                P0 += e * xv.x;  P1 += e * xv.y;
                P2 += e * xv.z;  P3 += e * xv.w;
            }
        }
        // cross-lane merge of (mx, S, P)
        for (int off = 16; off > 0; off >>= 1) {
            float mo = __shfl_xor(mx, off);
            float So = __shfl_xor(S, off);
            float Q0 = __shfl_xor(P0, off);
            float Q1 = __shfl_xor(P1, off);
            float Q2 = __shfl_xor(P2, off);
            float Q3 = __shfl_xor(P3, off);
            float nm = fmaxf(mx, mo);
            float ea = expf(mx - nm);
            float eb = expf(mo - nm);
            S  = S  * ea + So * eb;
            P0 = P0 * ea + Q0 * eb;  P1 = P1 * ea + Q1 * eb;
            P2 = P2 * ea + Q2 * eb;  P3 = P3 * ea + Q3 * eb;
            mx = nm;
        }
        if (lane == 0) {
            float inv = 1.f / S;
            s_xbar[wave][0] = P0 * inv;  s_xbar[wave][1] = P1 * inv;
            s_xbar[wave][2] = P2 * inv;  s_xbar[wave][3] = P3 * inv;
        }
    }
    __syncthreads();

    // ---------------- Phase 2: WMMA chain xbar -> hbar -> pooled -> feat ------
    // Tiles: M = 16 batches, N tile nt (16 cols) owned by wave nt (waves 0..3).
    const int lh = lane & 15;       // col (B/C/D) or row (A)
    const int hi = lane >> 4;       // half select

    // 2a: hbar(16x64) = xbar(16x4) @ W_in(4x64) + b_in  (single K=4 WMMA/tile)
    if (wave < 4) {
        const int nt = wave;
        v2f a, bm;
        a.x  = s_xbar[lh][hi * 2 + 0];
        a.y  = s_xbar[lh][hi * 2 + 1];
        bm.x = W_in[(hi * 2 + 0) * DDIM + nt * 16 + lh];
        bm.y = W_in[(hi * 2 + 1) * DDIM + nt * 16 + lh];
        float bias = b_in[nt * 16 + lh];
        v8f c = {bias, bias, bias, bias, bias, bias, bias, bias};
        c = __builtin_amdgcn_wmma_f32_16x16x4_f32(false, a, false, bm,
                                                  (short)0, c, false, false);
        #pragma unroll
        for (int v = 0; v < 8; ++v) s_hbar[v + hi * 8][nt * 16 + lh] = c[v];
    }
    __syncthreads();

    // 2b: pooled(16x64) = hbar @ Wv
    if (wave < 4) {
        const int nt = wave;
        v8f c = {0.f, 0.f, 0.f, 0.f, 0.f, 0.f, 0.f, 0.f};
        #pragma unroll 4
        for (int kk = 0; kk < DDIM; kk += 4) {
            v2f a, bm;
            a.x  = s_hbar[lh][kk + hi * 2 + 0];
            a.y  = s_hbar[lh][kk + hi * 2 + 1];
            bm.x = Wv[(kk + hi * 2 + 0) * DDIM + nt * 16 + lh];
            bm.y = Wv[(kk + hi * 2 + 1) * DDIM + nt * 16 + lh];
            c = __builtin_amdgcn_wmma_f32_16x16x4_f32(false, a, false, bm,
                                                      (short)0, c, false, false);
        }
        #pragma unroll
        for (int v = 0; v < 8; ++v) s_pool[v + hi * 8][nt * 16 + lh] = c[v];
    }
    __syncthreads();

    // 2c: feat(16x64) = pooled @ Wo + b_o
    if (wave < 4) {
        const int nt = wave;
        float bias = b_o[nt * 16 + lh];
        v8f c = {bias, bias, bias, bias, bias, bias, bias, bias};
        #pragma unroll 4
        for (int kk = 0; kk < DDIM; kk += 4) {
            v2f a, bm;
            a.x  = s_pool[lh][kk + hi * 2 + 0];
            a.y  = s_pool[lh][kk + hi * 2 + 1];
            bm.x = Wo[(kk + hi * 2 + 0) * DDIM + nt * 16 + lh];
            bm.y = Wo[(kk + hi * 2 + 1) * DDIM + nt * 16 + lh];
            c = __builtin_amdgcn_wmma_f32_16x16x4_f32(false, a, false, bm,
                                                      (short)0, c, false, false);
        }
        #pragma unroll
        for (int v = 0; v < 8; ++v) s_feat[v + hi * 8][nt * 16 + lh] = c[v];
    }
    __syncthreads();

    // ---------------- Phase 3: g[b] = feat . w_loc[4:] + b_loc ----------------
    {
        float gp = s_feat[wave][lane]      * w_loc[4 + lane]
                 + s_feat[wave][lane + 32] * w_loc[36 + lane];
        for (int off = 16; off > 0; off >>= 1) gp += __shfl_xor(gp, off);
        if (lane == 0) s_g[wave] = gp + b_loc[0];
    }
    __syncthreads();

    // ---------------- Phase 4: weighted 4x4 normal equations + solve ----------
    {
        const float wl0 = w_loc[0], wl1 = w_loc[1], wl2 = w_loc[2], wl3 = w_loc[3];
        const float g = s_g[wave];
        const float ap = prelu_a[0];
        const float4* xb = reinterpret_cast<const float4*>(x + (size_t)b * MTOK * 4);
        const unsigned char* pm = pad + (size_t)b * MTOK;

        float G00=0,G01=0,G02=0,G03=0,G11=0,G12=0,G13=0,G22=0,G23=0,G33=0;
        float c0=0,c1=0,c2=0,c3=0;

        for (int m = lane; m < MTOK; m += 32) {
            float4 xv = xb[m];
            float z = xv.x * wl0 + xv.y * wl1 + xv.z * wl2 + xv.w * wl3 + g;
            float w = (z >= 0.f) ? z : ap * z;
            if (pm[m]) w = 0.f;
            float w2 = w * w;
            float a0 = -xv.y, a1 = -xv.z, a2 = -xv.w;  // a3 = 1
            float r = xv.x;
            G00 += w2 * a0 * a0;  G01 += w2 * a0 * a1;  G02 += w2 * a0 * a2;
            G03 += w2 * a0;       G11 += w2 * a1 * a1;  G12 += w2 * a1 * a2;
            G13 += w2 * a1;       G22 += w2 * a2 * a2;  G23 += w2 * a2;
            G33 += w2;
            c0 += w2 * a0 * r;  c1 += w2 * a1 * r;  c2 += w2 * a2 * r;  c3 += w2 * r;
        }
        for (int off = 16; off > 0; off >>= 1) {
            G00 += __shfl_xor(G00, off);  G01 += __shfl_xor(G01, off);
            G02 += __shfl_xor(G02, off);  G03 += __shfl_xor(G03, off);
            G11 += __shfl_xor(G11, off);  G12 += __shfl_xor(G12, off);
            G13 += __shfl_xor(G13, off);  G22 += __shfl_xor(G22, off);
            G23 += __shfl_xor(G23, off);  G33 += __shfl_xor(G33, off);
            c0  += __shfl_xor(c0,  off);  c1  += __shfl_xor(c1,  off);
            c2  += __shfl_xor(c2,  off);  c3  += __shfl_xor(c3,  off);
        }
        if (lane == 0) {
            float A[4][5] = {
                {G00, G01, G02, G03, c0},
                {G01, G11, G12, G13, c1},
                {G02, G12, G22, G23, c2},
                {G03, G13, G23, G33, c3}};
            // Gaussian elimination with partial pivoting
            for (int col = 0; col < 4; ++col) {
                int piv = col; float best = fabsf(A[col][col]);
                for (int rr = col + 1; rr < 4; ++rr) {
                    float v = fabsf(A[rr][col]);
                    if (v > best) { best = v; piv = rr; }
                }
                if (piv != col) {
                    for (int j = col; j < 5; ++j) {
                        float t = A[col][j]; A[col][j] = A[piv][j]; A[piv][j] = t;
                    }
                }
                float d = A[col][col];
                float inv = (fabsf(d) > 1e-30f) ? 1.f / d : 0.f;
                for (int rr = col + 1; rr < 4; ++rr) {
                    float f = A[rr][col] * inv;
                    for (int j = col; j < 5; ++j) A[rr][j] -= f * A[col][j];
                }
            }
            float beta[4];
            for (int rr = 3; rr >= 0; --rr) {
                float acc = A[rr][4];
                for (int j = rr + 1; j < 4; ++j) acc -= A[rr][j] * beta[j];
                float d = A[rr][rr];
                beta[rr] = (fabsf(d) > 1e-30f) ? acc / d : 0.f;
            }
            out[(size_t)b * 4 + 0] = beta[0];
            out[(size_t)b * 4 + 1] = beta[1];
            out[(size_t)b * 4 + 2] = beta[2];
            out[(size_t)b * 4 + 3] = beta[3];
        }
    }
}

extern "C" void kernel_launch(void* const* d_in, const int* in_sizes, int n_in,
                              void* d_out, int out_size, void* d_ws, size_t ws_size,
                              hipStream_t stream) {
    (void)in_sizes; (void)n_in; (void)out_size; (void)d_ws; (void)ws_size;
    const float*         x      = (const float*)d_in[0];
    const unsigned char* pad    = (const unsigned char*)d_in[1];
    const float*         W_in   = (const float*)d_in[2];
    const float*         b_in   = (const float*)d_in[3];
    const float*         seed   = (const float*)d_in[4];
    const float*         Wq     = (const float*)d_in[5];
    const float*         Wk     = (const float*)d_in[6];
    const float*         Wv     = (const float*)d_in[7];
    const float*         Wo     = (const float*)d_in[8];
    const float*         b_o    = (const float*)d_in[9];
    const float*         w_loc  = (const float*)d_in[10];
    const float*         b_loc  = (const float*)d_in[11];
    const float*         prelua = (const float*)d_in[12];
    float* out = (float*)d_out;

    dim3 grid(BDIM / TB);   // 128 workgroups, 16 batches each
    dim3 block(512);        // 16 wave32 waves: one wave per batch
    hipLaunchKernelGGL(wls_settf_fused, grid, block, 0, stream,
                       x, pad, W_in, b_in, seed, Wq, Wk, Wv, Wo, b_o,
                       w_loc, b_loc, prelua, out);
}